// _CARAFE2d_16595753631966
// MI455X (gfx1250) — compile-verified
//
#include <hip/hip_runtime.h>
#include <math.h>

typedef __attribute__((ext_vector_type(2))) float v2f;
typedef __attribute__((ext_vector_type(8))) float v8f;

#define CC   256   // input channels
#define MM   64    // compressed channels
#define KK   25    // kernel k*k
#define JJ   100   // kk * s * s
#define HH   64
#define WW   64
#define HWSZ 4096
#define BB   4

// ---------------------------------------------------------------------------
// Kernel 1: fused  [compress 1x1 conv GEMM -> relu] -> [predict GEMM + bias]
//           -> [softmax over kk]  using V_WMMA_F32_16X16X4_F32.
// One block (128 thr = 4 waves) handles a 16-pixel tile.
// Wave wv computes hid rows [wv*16, wv*16+16) for all 16 pixels (K=256 loop),
// stores relu(hid) to LDS, then waves cover the 7 j-tiles of the predict GEMM
// (j padded 100->112) reading hid from LDS. Softmax result -> workspace.
// ---------------------------------------------------------------------------
__global__ void __launch_bounds__(128)
carafe_weights_kernel(const float* __restrict__ x,
                      const float* __restrict__ comp_w,
                      const float* __restrict__ comp_b,
                      const float* __restrict__ pred_w,
                      const float* __restrict__ pred_b,
                      float* __restrict__ wsm)
{
    __shared__ float hid_lds[MM * 17];   // hid[m][pixcol], stride 17 (no bank conflicts)
    __shared__ float w_lds[16 * 113];    // w[pixcol][j],  stride 113 (odd)

    const int tid  = threadIdx.x;
    const int wv   = tid >> 5;       // wave id 0..3 (wave32)
    const int lane = tid & 31;
    const int lrow = lane & 15;      // M/N row within fragment
    const int lhi  = lane >> 4;      // 0: K=0,1 half   1: K=2,3 half

    const int n0  = blockIdx.x * 16;         // first pixel of tile
    const int pix = n0 + lrow;               // this lane's pixel column
    const int b   = pix >> 12;               // HW = 4096
    const int hw  = pix & (HWSZ - 1);
    const float* xb = x + (size_t)b * CC * HWSZ + hw;

    // ---- GEMM1: hid[m,pix] = sum_c comp_w[m,c] * x[c,pix] -----------------
    const int m0 = wv * 16;
    v8f acc = {};
    for (int k0 = 0; k0 < CC; k0 += 4) {
        const int ka = k0 + 2 * lhi;
        v2f a, bf;
        // A frag: 16x4 fp32, lane holds A[M=lrow, K=ka..ka+1]
        a.x = comp_w[(m0 + lrow) * CC + ka];
        a.y = comp_w[(m0 + lrow) * CC + ka + 1];
        // B frag: 4x16 fp32, lane holds B[K=ka..ka+1, N=lrow]
        bf.x = xb[(size_t)ka * HWSZ];
        bf.y = xb[(size_t)(ka + 1) * HWSZ];
        acc = __builtin_amdgcn_wmma_f32_16x16x4_f32(
            false, a, false, bf, (short)0, acc, false, false);
    }
    // D layout: VGPR v holds D[M = v + 8*lhi, N = lrow]
#pragma unroll
    for (int v = 0; v < 8; ++v) {
        int m = m0 + v + 8 * lhi;
        float h = acc[v] + comp_b[m];
        hid_lds[m * 17 + lrow] = h > 0.f ? h : 0.f;   // relu
    }
    __syncthreads();

    // ---- GEMM2: w[j,pix] = sum_m pred_w[j,m] * hid[m,pix] + pred_b[j] -----
    // 7 j-tiles of 16 (rows 100..111 padded with zeros). Wave-uniform guard.
    for (int tt = 0; tt < 2; ++tt) {
        const int t = wv + 4 * tt;
        if (t < 7) {
            v8f acc2 = {};
            const int ja = t * 16 + lrow;
            const bool jv = (ja < JJ);
            for (int k0 = 0; k0 < MM; k0 += 4) {
                const int ka = k0 + 2 * lhi;
                v2f a, bf;
                a.x = jv ? pred_w[ja * MM + ka]     : 0.f;
                a.y = jv ? pred_w[ja * MM + ka + 1] : 0.f;
                bf.x = hid_lds[ka * 17 + lrow];
                bf.y = hid_lds[(ka + 1) * 17 + lrow];
                acc2 = __builtin_amdgcn_wmma_f32_16x16x4_f32(
                    false, a, false, bf, (short)0, acc2, false, false);
            }
#pragma unroll
            for (int v = 0; v < 8; ++v) {
                int j = t * 16 + v + 8 * lhi;
                if (j < JJ)
                    w_lds[lrow * 113 + j] = acc2[v] + pred_b[j];
            }
        }
    }
    __syncthreads();

    // ---- softmax over kk=25 for each (pixel, pq) --------------------------
    // j = k*4 + p*2 + q  (pixel-shuffle layout). 64 tasks -> waves 0,1.
    if (tid < 64) {
        const int pcol = tid >> 2;
        const int pq   = tid & 3;
        const float* row = &w_lds[pcol * 113 + pq];
        float vals[KK];
        float mx = -1e30f;
#pragma unroll
        for (int k = 0; k < KK; ++k) {
            vals[k] = row[k * 4];
            mx = fmaxf(mx, vals[k]);
        }
        float s = 0.f;
#pragma unroll
        for (int k = 0; k < KK; ++k) {
            vals[k] = __expf(vals[k] - mx);
            s += vals[k];
        }
        const float inv = 1.f / s;
        const int gpix = n0 + pcol;
        float* outp = wsm + (size_t)gpix * JJ + pq;
#pragma unroll
        for (int k = 0; k < KK; ++k)
            outp[k * 4] = vals[k] * inv;
    }
}

// ---------------------------------------------------------------------------
// Kernel 2: gather / reassembly (memory-bound stage).
// One block per (b, h) row: stage 64 pixels x 100 weights in LDS (stride 104
// -> 16B-aligned float4 reads), then loop channels 4 at a time. Thread (c,w)
// reads its 5x5 zero-padded patch (coalesced in w) and emits 4 outputs as two
// contiguous float2 stores (threads tile the 128-wide output row exactly).
// ---------------------------------------------------------------------------
__global__ void __launch_bounds__(256)
carafe_gather_kernel(const float* __restrict__ x,
                     const float* __restrict__ wsm,
                     float* __restrict__ out)
{
    __shared__ __align__(16) float wlds[64 * 104];

    const int bh = blockIdx.x;      // 0..B*H-1
    const int b  = bh >> 6;
    const int h  = bh & 63;
    const int tid = threadIdx.x;

    // stage this row's softmax weights
    const size_t pixbase = (size_t)(b * HWSZ + h * WW);
    for (int idx = tid; idx < 64 * JJ; idx += 256) {
        int pw = idx / JJ;
        int j  = idx - pw * JJ;
        wlds[pw * 104 + j] = wsm[(pixbase + pw) * JJ + j];
    }
    __syncthreads();

    const int w    = tid & 63;
    const int csub = tid >> 6;
    const float4* wt = (const float4*)&wlds[w * 104];

    for (int cg = 0; cg < CC; cg += 4) {
        const int c = cg + csub;
        const float* xrow = x + (size_t)(b * CC + c) * HH * WW;
        float a0 = 0.f, a1 = 0.f, a2 = 0.f, a3 = 0.f;
        int k = 0;
#pragma unroll
        for (int di = 0; di < 5; ++di) {
            const int hh2 = h + di - 2;
            const bool hok = (unsigned)hh2 < (unsigned)HH;
#pragma unroll
            for (int dj = 0; dj < 5; ++dj, ++k) {
                const int wj = w + dj - 2;
                const float xv =
                    (hok && (unsigned)wj < (unsigned)WW) ? xrow[hh2 * WW + wj] : 0.f;
                const float4 w4 = wt[k];            // j = k*4 + {0,1,2,3}
                a0 += xv * w4.x;                    // (p=0,q=0)
                a1 += xv * w4.y;                    // (p=0,q=1)
                a2 += xv * w4.z;                    // (p=1,q=0)
                a3 += xv * w4.w;                    // (p=1,q=1)
            }
        }
        float* orow = out + ((size_t)(b * CC + c) * (2 * HH) + 2 * h) * (2 * WW) + 2 * w;
        *(float2*)(orow)            = make_float2(a0, a1);
        *(float2*)(orow + 2 * WW)   = make_float2(a2, a3);
    }
}

// ---------------------------------------------------------------------------
extern "C" void kernel_launch(void* const* d_in, const int* in_sizes, int n_in,
                              void* d_out, int out_size, void* d_ws, size_t ws_size,
                              hipStream_t stream)
{
    (void)in_sizes; (void)n_in; (void)out_size; (void)ws_size;
    const float* x      = (const float*)d_in[0];
    const float* comp_w = (const float*)d_in[1];
    const float* comp_b = (const float*)d_in[2];
    const float* pred_w = (const float*)d_in[3];
    const float* pred_b = (const float*)d_in[4];
    float* out = (float*)d_out;
    float* wsm = (float*)d_ws;   // 16384 * 100 floats = 6.55 MB of scratch

    // 16384 pixels / 16 per block
    carafe_weights_kernel<<<1024, 128, 0, stream>>>(x, comp_w, comp_b,
                                                    pred_w, pred_b, wsm);
    // one block per (b, h) output-source row
    carafe_gather_kernel<<<BB * HH, 256, 0, stream>>>(x, wsm, out);
}